// RelationNetwork_63917703299558
// MI455X (gfx1250) — compile-verified
//
#include <hip/hip_runtime.h>

// CDNA5 / gfx1250 Relation Network, fp32 end-to-end using V_WMMA_F32_16X16X4_F32.
//
// Reference decomposition (same as the JAX reference):
//   a = x @ gW1[:256], b = x @ gW1[256:512], cb = q @ gW1[512:] + gb1
//   h1[pair(j,i)] = relu(a[i] + b[j] + cb)
//   h2 = relu(h1 @ gW2 + gb2); rel = relu(h2 @ gW3 + gb3)
//   emb = sum over 512*512 pairs; then small MLP f.
//
// Main kernel: block = 1 j x 128 i's. Waves split the N (output-feature)
// dimension: wave w owns n-tiles {2w, 2w+1} for all 128 rows, so each pair of
// B-fragment loads feeds 8 WMMAs (8 M-tiles). Each block reads the 512 KB of
// swizzled weights exactly once from L2.

typedef __attribute__((ext_vector_type(2))) float v2f;
typedef __attribute__((ext_vector_type(8))) float v8f;

#define NF 512
#define DD 256
#define HGW 256
#define HPAD 260   // LDS row pitch in floats (pad 4 -> conflict-free frag reads)

// ---------------------------------------------------------------- prep kernels

__global__ void prep_ab_kernel(const float* __restrict__ x,
                               const float* __restrict__ gW1,
                               float* __restrict__ a, float* __restrict__ b) {
    __shared__ float xr[DD];
    const int i = blockIdx.x;          // fact row
    const int f = threadIdx.x;         // output feature
    xr[f] = x[i * DD + f];
    __syncthreads();
    float sa = 0.f, sb = 0.f;
    for (int k = 0; k < DD; ++k) {
        const float xv = xr[k];
        sa += xv * gW1[k * HGW + f];            // top block of gW1
        sb += xv * gW1[(DD + k) * HGW + f];     // middle block
    }
    a[i * HGW + f] = sa;
    b[i * HGW + f] = sb;
}

__global__ void prep_cb_kernel(const float* __restrict__ q,
                               const float* __restrict__ gW1,
                               const float* __restrict__ gb1,
                               float* __restrict__ cb, float* __restrict__ emb) {
    const int f = threadIdx.x;
    float s = gb1[f];
    for (int k = 0; k < DD; ++k) s += q[k] * gW1[(2 * DD + k) * HGW + f];
    cb[f] = s;
    emb[f] = 0.f;   // zero the global pair-sum accumulator every call
}

// Swizzle W2/W3 into WMMA f32 16x16x4 B-fragment order:
//   v2f index r = kc*512 + nt*32 + lane,
//   lane<16 :  { W[4kc+0][16nt+lane],    W[4kc+1][16nt+lane]    }
//   lane>=16:  { W[4kc+2][16nt+lane-16], W[4kc+3][16nt+lane-16] }
__global__ void swizzle_w_kernel(const float* __restrict__ W2,
                                 const float* __restrict__ W3,
                                 float* __restrict__ W2f, float* __restrict__ W3f) {
    const int g = blockIdx.x * blockDim.x + threadIdx.x;   // 0..131071
    const int m    = g >> 16;
    const int r    = g & 0xFFFF;
    const int lane = r & 31;
    const int nt   = (r >> 5) & 15;
    const int kc   = r >> 9;
    const int koff = (lane >> 4) << 1;
    const int col  = nt * 16 + (lane & 15);
    const int k    = kc * 4 + koff;
    const float* src = m ? W3 : W2;
    float*       dst = m ? W3f : W2f;
    dst[2 * r]     = src[k * HGW + col];
    dst[2 * r + 1] = src[(k + 1) * HGW + col];
}

// ---------------------------------------------------------------- main kernel
// grid = 2048 blocks: j = blk>>2, i0 = (blk&3)*128. 256 threads = 8 waves.

__global__ void __launch_bounds__(256)
relnet_pair_kernel(const float* __restrict__ a, const float* __restrict__ b,
                   const float* __restrict__ cb,
                   const float* __restrict__ W2f, const float* __restrict__ gb2,
                   const float* __restrict__ W3f, const float* __restrict__ gb3,
                   float* __restrict__ emb) {
    __shared__ float Hs[128 * HPAD];   // 130 KB activation tile (CDNA5: 320KB/WGP)
    __shared__ float part[HGW];        // per-block partial pair-sum

    const int tid  = threadIdx.x;
    const int w    = tid >> 5;
    const int lane = tid & 31;
    const int j    = blockIdx.x >> 2;
    const int i0   = (blockIdx.x & 3) * 128;

    part[tid] = 0.f;

    // ---- phase 1: fill H1 rows 16w..16w+15: relu(a[i] + b[j] + cb) ----
    float bc[8];
#pragma unroll
    for (int t = 0; t < 8; ++t) {
        const int col = lane + 32 * t;
        bc[t] = b[j * HGW + col] + cb[col];
    }
    for (int r = 0; r < 16; ++r) {
        const int row = 16 * w + r;
        const float* ar = a + (size_t)(i0 + row) * HGW;
#pragma unroll
        for (int t = 0; t < 8; ++t) {
            const int col = lane + 32 * t;
            Hs[row * HPAD + col] = fmaxf(ar[col] + bc[t], 0.f);
        }
    }
    __syncthreads();

    // WMMA f32 16x16x4 fragment addressing
    const int mrow = lane & 15;           // A-frag: row within M-tile
    const int koff = (lane >> 4) << 1;    // A-frag: lane<16 -> K 0..1, else K 2..3
    const int colb = lane & 15;           // C/D col within N-tile
    const v8f vzero = {};

    // ---- layer 2: H2 = relu(H1 @ W2 + gb2); wave owns n-tiles 2w,2w+1 ----
    {
        v8f acc[8][2];
#pragma unroll
        for (int mt = 0; mt < 8; ++mt) { acc[mt][0] = vzero; acc[mt][1] = vzero; }
        for (int kc = 0; kc < 64; ++kc) {
            const v2f* wp = (const v2f*)W2f + (size_t)kc * 512 + w * 64 + lane;
            const v2f bf0 = wp[0];    // n-tile 2w
            const v2f bf1 = wp[32];   // n-tile 2w+1
            const int kbase = 4 * kc + koff;
#pragma unroll
            for (int mt = 0; mt < 8; ++mt) {
                const v2f af = *(const v2f*)&Hs[(16 * mt + mrow) * HPAD + kbase];
                acc[mt][0] = __builtin_amdgcn_wmma_f32_16x16x4_f32(
                    false, af, false, bf0, (short)0, acc[mt][0], false, false);
                acc[mt][1] = __builtin_amdgcn_wmma_f32_16x16x4_f32(
                    false, af, false, bf1, (short)0, acc[mt][1], false, false);
            }
        }
        __syncthreads();   // every wave done reading H1 -> safe to overwrite
#pragma unroll
        for (int ntl = 0; ntl < 2; ++ntl) {
            const int nt = 2 * w + ntl;
            const float bias = gb2[nt * 16 + colb];
#pragma unroll
            for (int mt = 0; mt < 8; ++mt) {
#pragma unroll
                for (int r = 0; r < 8; ++r) {
                    const int row = 16 * mt + r + ((lane >> 4) << 3);  // C/D layout
                    Hs[row * HPAD + nt * 16 + colb] = fmaxf(acc[mt][ntl][r] + bias, 0.f);
                }
            }
        }
    }
    __syncthreads();       // H2 complete

    // ---- layer 3 + reduction over pairs ----
    {
        v8f acc[8][2];
#pragma unroll
        for (int mt = 0; mt < 8; ++mt) { acc[mt][0] = vzero; acc[mt][1] = vzero; }
        for (int kc = 0; kc < 64; ++kc) {
            const v2f* wp = (const v2f*)W3f + (size_t)kc * 512 + w * 64 + lane;
            const v2f bf0 = wp[0];
            const v2f bf1 = wp[32];
            const int kbase = 4 * kc + koff;
#pragma unroll
            for (int mt = 0; mt < 8; ++mt) {
                const v2f af = *(const v2f*)&Hs[(16 * mt + mrow) * HPAD + kbase];
                acc[mt][0] = __builtin_amdgcn_wmma_f32_16x16x4_f32(
                    false, af, false, bf0, (short)0, acc[mt][0], false, false);
                acc[mt][1] = __builtin_amdgcn_wmma_f32_16x16x4_f32(
                    false, af, false, bf1, (short)0, acc[mt][1], false, false);
            }
        }
#pragma unroll
        for (int ntl = 0; ntl < 2; ++ntl) {
            const int nt = 2 * w + ntl;
            const float bias = gb3[nt * 16 + colb];
            float s = 0.f;
#pragma unroll
            for (int mt = 0; mt < 8; ++mt) {
#pragma unroll
                for (int r = 0; r < 8; ++r) s += fmaxf(acc[mt][ntl][r] + bias, 0.f);
            }
            atomicAdd(&part[nt * 16 + colb], s);   // ds_add_f32 (2-way conflict max)
        }
    }
    __syncthreads();
    atomicAdd(&emb[tid], part[tid]);               // global_atomic_add_f32
}

// ---------------------------------------------------------------- final MLP f

__global__ void final_mlp_kernel(const float* __restrict__ emb,
                                 const float* __restrict__ fW1, const float* __restrict__ fb1,
                                 const float* __restrict__ fW2, const float* __restrict__ fb2,
                                 const float* __restrict__ fW3, const float* __restrict__ fb3,
                                 float* __restrict__ out) {
    __shared__ float e[256];
    __shared__ float y1[256];
    __shared__ float y2[512];
    const int t = threadIdx.x;   // 512 threads
    if (t < 256) e[t] = emb[t];
    __syncthreads();
    if (t < 256) {
        float s = fb1[t];
        for (int k = 0; k < 256; ++k) s += e[k] * fW1[k * 256 + t];
        y1[t] = fmaxf(s, 0.f);
    }
    __syncthreads();
    {
        float s = fb2[t];
        for (int k = 0; k < 256; ++k) s += y1[k] * fW2[k * 512 + t];
        y2[t] = fmaxf(s, 0.f);
    }
    __syncthreads();
    if (t < 159) {
        float s = fb3[t];
        for (int k = 0; k < 512; ++k) s += y2[k] * fW3[k * 159 + t];
        out[t] = s;
    }
}

// ---------------------------------------------------------------- launch

extern "C" void kernel_launch(void* const* d_in, const int* in_sizes, int n_in,
                              void* d_out, int out_size, void* d_ws, size_t ws_size,
                              hipStream_t stream) {
    const float* x   = (const float*)d_in[0];
    const float* q   = (const float*)d_in[1];
    const float* gW1 = (const float*)d_in[2];
    const float* gb1 = (const float*)d_in[3];
    const float* gW2 = (const float*)d_in[4];
    const float* gb2 = (const float*)d_in[5];
    const float* gW3 = (const float*)d_in[6];
    const float* gb3 = (const float*)d_in[7];
    const float* fW1 = (const float*)d_in[8];
    const float* fb1 = (const float*)d_in[9];
    const float* fW2 = (const float*)d_in[10];
    const float* fb2 = (const float*)d_in[11];
    const float* fW3 = (const float*)d_in[12];
    const float* fb3 = (const float*)d_in[13];

    // workspace layout (floats); total 393728 floats = ~1.5 MB
    float* ws  = (float*)d_ws;
    float* a   = ws;                     // 512*256
    float* b   = a + NF * HGW;           // 512*256
    float* cb  = b + NF * HGW;           // 256
    float* emb = cb + HGW;               // 256
    float* W2f = emb + HGW;              // 65536
    float* W3f = W2f + 64 * 16 * 32 * 2; // 65536
    float* out = (float*)d_out;

    prep_ab_kernel<<<NF, 256, 0, stream>>>(x, gW1, a, b);
    prep_cb_kernel<<<1, 256, 0, stream>>>(q, gW1, gb1, cb, emb);
    swizzle_w_kernel<<<512, 256, 0, stream>>>(gW2, gW3, W2f, W3f);
    relnet_pair_kernel<<<2048, 256, 0, stream>>>(a, b, cb, W2f, gb2, W3f, gb3, emb);
    final_mlp_kernel<<<1, 512, 0, stream>>>(emb, fW1, fb1, fW2, fb2, fW3, fb3, out);
}